// MeanPoolingWithoutPadding_15453292331085
// MI455X (gfx1250) — compile-verified
//
#include <hip/hip_runtime.h>

// Problem constants from the reference: B=64, S=2048, D=512, fp32.
#define BB 64
#define SS 2048
#define DD 512
#define MAX_SPLIT 16

typedef __attribute__((ext_vector_type(2))) float v2f;
typedef __attribute__((ext_vector_type(8))) float v8f;

__device__ __forceinline__ int imin(int a, int b) { return a < b ? a : b; }

// One wave owns a 16-column d-tile of one batch and streams an S-chunk,
// accumulating with V_WMMA_F32_16X16X4_F32:
//   A (16x4)  = mask bits for the 4 rows, replicated over M (register constant)
//   B (4x16)  = features tile, lanes 0-15 hold row K, lanes 16-31 row K+2
//   C (16x16) = every M-row equals the 16 masked column sums
__global__ void __launch_bounds__(128)
mean_pool_partial(const float* __restrict__ feat,
                  const int*   __restrict__ lengths,
                  float*       __restrict__ partial,   // [split][B][D]
                  int chunk)
{
    const int lane  = threadIdx.x & 31;
    const int wave  = threadIdx.x >> 5;                // 0..3
    const int dtile = blockIdx.x * 4 + wave;           // 0..31
    const int b     = blockIdx.y;                      // 0..63
    const int split = blockIdx.z;
    const int d0    = dtile * 16;

    const int len = lengths[b];
    const int s0  = split * chunk;
    const int s1  = imin(s0 + chunk, len);             // skip masked-out rows entirely

    const float* __restrict__ base = feat + (size_t)b * SS * DD;

    const int col = lane & 15;                         // N index
    const int khi = (lane >> 4) << 1;                  // 0 (lanes 0-15) or 2 (lanes 16-31)

    // Per-lane streaming pointer: row (s0+khi), column (d0+col).
    // Only dereferenced when the loops run, so it's safe even if s0 >= len.
    const float* __restrict__ p =
        base + (size_t)(s0 + khi) * DD + d0 + col;

    v8f c = {0.f, 0.f, 0.f, 0.f, 0.f, 0.f, 0.f, 0.f};
    const v2f a_ones = {1.0f, 1.0f};

    int s = s0;
    // Hot loop: 16 rows per block = 4 WMMAs, 8 loads with immediate offsets,
    // one pointer increment.
    for (; s + 16 <= s1; s += 16) {
        #pragma unroll
        for (int j = 0; j < 4; ++j) {
            v2f bm;
            bm.x = p[j * 4 * DD];
            bm.y = p[j * 4 * DD + DD];
            c = __builtin_amdgcn_wmma_f32_16x16x4_f32(
                    /*neg_a=*/false, a_ones, /*neg_b=*/false, bm,
                    /*c_mod=*/(short)0, c, /*reuse_a=*/false, /*reuse_b=*/false);
        }
        p += 16 * DD;
    }
    // Remainder full K=4 steps.
    for (; s + 4 <= s1; s += 4) {
        v2f bm;
        bm.x = p[0];
        bm.y = p[DD];
        c = __builtin_amdgcn_wmma_f32_16x16x4_f32(
                false, a_ones, false, bm, (short)0, c, false, false);
        p += 4 * DD;
    }
    // Tail (1..3 valid rows): zero the invalid K lanes of A, clamp row loads.
    if (s < s1) {
        const int rem = s1 - s;                        // 1..3
        v2f a, bm;
        a.x = (khi     < rem) ? 1.0f : 0.0f;
        a.y = (khi + 1 < rem) ? 1.0f : 0.0f;
        const int r0 = imin(khi,     rem - 1);         // safe in-bounds rows
        const int r1 = imin(khi + 1, rem - 1);
        bm.x = base[(size_t)(s + r0) * DD + d0 + col];
        bm.y = base[(size_t)(s + r1) * DD + d0 + col];
        c = __builtin_amdgcn_wmma_f32_16x16x4_f32(
                false, a, false, bm, (short)0, c, false, false);
    }

    // Row M=0 of C lives in c[0] on lanes 0-15 (N = lane). Always write,
    // even for empty chunks (zeros), so the partial buffer is fully defined.
    if (lane < 16) {
        partial[((size_t)split * BB + b) * DD + d0 + lane] = c[0];
    }
}

// Sum the SPLIT partial buffers and divide by length.
__global__ void __launch_bounds__(256)
mean_pool_finalize(const float* __restrict__ partial,
                   const int*   __restrict__ lengths,
                   float*       __restrict__ out,
                   int split_cnt)
{
    const int idx = blockIdx.x * 256 + threadIdx.x;    // over B*D
    if (idx >= BB * DD) return;
    const int b = idx / DD;
    float sum = 0.f;
    for (int sp = 0; sp < split_cnt; ++sp)
        sum += partial[(size_t)sp * BB * DD + idx];
    out[idx] = sum / (float)lengths[b];
}

extern "C" void kernel_launch(void* const* d_in, const int* in_sizes, int n_in,
                              void* d_out, int out_size, void* d_ws, size_t ws_size,
                              hipStream_t stream)
{
    const float* feat    = (const float*)d_in[0];
    const int*   lengths = (const int*)d_in[1];
    float*       out     = (float*)d_out;

    // Choose the S-split factor by available workspace (deterministic partials,
    // no float atomics). 16 splits -> 2 MiB of fp32 partials.
    int split = (int)(ws_size / ((size_t)BB * DD * sizeof(float)));
    if (split > MAX_SPLIT) split = MAX_SPLIT;
    float* partial = (float*)d_ws;
    if (split < 1) {                 // degenerate: no usable workspace
        split = 1;
        partial = out;               // finalize then rescales in place
    }
    const int chunk = (SS + split - 1) / split;

    dim3 grid1(DD / 16 / 4, BB, split);  // 8 x 64 x split blocks, 4 waves each
    mean_pool_partial<<<grid1, 128, 0, stream>>>(feat, lengths, partial, chunk);

    const int total = BB * DD;
    mean_pool_finalize<<<(total + 255) / 256, 256, 0, stream>>>(partial, lengths, out, split);
}